// TargetCriterion_86165633892924
// MI455X (gfx1250) — compile-verified
//
#include <hip/hip_runtime.h>
#include <hip/hip_bf16.h>

// ---------------- problem constants (match reference) ----------------
#define B_     32
#define N_     8400
#define C_     80
#define SIDES_ 4
#define BINS_  17
#define NC_    (N_ * C_)   // 672000
#define NC4_   (NC_ / 4)   // 168000 float4 per batch
#define N4_    (N_ / 4)    // 2100 vec4 anchors per side
#define QCH_   96          // QFL chunks (blocks) per batch
#define DCH_   33          // DFL chunks (blocks) per batch (4*N/4 = 8400 slots)
#define EPS_   0.001f

typedef float v2f __attribute__((ext_vector_type(2)));
typedef float v4f __attribute__((ext_vector_type(4)));
typedef float v8f __attribute__((ext_vector_type(8)));

// -------- wave32 reduction via V_WMMA_F32_16X16X4_F32 --------
// A = per-lane partials (16x4 f32: lane m holds A[m][0..1], lane m+16 holds
// A[m][2..3]), B = ones(4x16).  D[m][n] = sum_k A[m][k] = group sum g_m,
// replicated across n.  D VGPR r: lanes 0-15 hold g_r, lanes 16-31 hold
// g_{8+r}.  Sum the 8 D registers per lane, then one cross-half shuffle.
__device__ __forceinline__ float wave_reduce_wmma(float p0, float p1) {
  v2f a; a[0] = p0;   a[1] = p1;
  v2f b; b[0] = 1.0f; b[1] = 1.0f;
  v8f c = {};
  c = __builtin_amdgcn_wmma_f32_16x16x4_f32(
      /*neg_a=*/false, a, /*neg_b=*/false, b,
      /*c_mod=*/(short)0, c, /*reuse_a=*/false, /*reuse_b=*/false);
  float t = ((c[0] + c[1]) + (c[2] + c[3])) + ((c[4] + c[5]) + (c[6] + c[7]));
  t += __shfl_xor(t, 16, 32);   // combine lane-halves (g_0..7 with g_8..15)
  return t;                     // full 32-lane sum of (p0 + p1), all lanes
}

// block of 256 = 8 wave32; result valid on thread 0
__device__ __forceinline__ float block_reduce_256(float p, float* sred) {
  float t = wave_reduce_wmma(p, 0.0f);
  const int wave = threadIdx.x >> 5;
  const int lane = threadIdx.x & 31;
  if (lane == 0) sred[wave] = t;
  __syncthreads();
  float s = 0.0f;
  if (threadIdx.x == 0) {
#pragma unroll
    for (int w = 0; w < 8; ++w) s += sred[w];
  }
  return s;
}

// ---------------- kernel 1: pos_sum[b] ----------------
__global__ void __launch_bounds__(256)
pos_sum_kernel(const unsigned char* __restrict__ pos_mask, float* __restrict__ ws_pos) {
  __shared__ float sred[8];
  const int b = blockIdx.x;
  const unsigned char* __restrict__ m = pos_mask + (size_t)b * N_;
  float acc = 0.0f;
  for (int n = threadIdx.x; n < N_; n += 256) acc += (float)m[n];
  float s = block_reduce_256(acc, sred);
  if (threadIdx.x == 0) ws_pos[b] = s;
}

// ---------------- kernel 2: QFL partial sums ----------------
__device__ __forceinline__ float qfl_elem(float x, float t, float m) {
  float e   = __expf(-fabsf(x));                 // exp(-|x|)
  float bce = fmaxf(x, 0.0f) - x * t + __logf(1.0f + e);
  float inv = 1.0f / (1.0f + e);                 // sigmoid(|x|)
  float sig = (x >= 0.0f) ? inv : (1.0f - inv);  // e/(1+e) = 1 - 1/(1+e)
  float d   = sig - t;
  return bce * d * d * m;
}

__global__ void __launch_bounds__(256)
qfl_partial_kernel(const float* __restrict__ pred_cls,
                   const float* __restrict__ pseudo_qfl,
                   const float* __restrict__ total_mask,
                   float* __restrict__ ws_qfl) {
  __shared__ float sred[8];
  const int b = blockIdx.y;
  const v4f* __restrict__ x4 = reinterpret_cast<const v4f*>(pred_cls   + (size_t)b * NC_);
  const v4f* __restrict__ t4 = reinterpret_cast<const v4f*>(pseudo_qfl + (size_t)b * NC_);
  const v4f* __restrict__ m4 = reinterpret_cast<const v4f*>(total_mask + (size_t)b * NC_);
  float acc = 0.0f;
  for (int v = blockIdx.x * 256 + threadIdx.x; v < NC4_; v += QCH_ * 256) {
    v4f x = __builtin_nontemporal_load(x4 + v);   // streamed once: TH=NT
    v4f t = __builtin_nontemporal_load(t4 + v);
    v4f m = __builtin_nontemporal_load(m4 + v);
#pragma unroll
    for (int j = 0; j < 4; ++j) acc += qfl_elem(x[j], t[j], m[j]);
  }
  float s = block_reduce_256(acc, sred);
  if (threadIdx.x == 0) ws_qfl[b * QCH_ + blockIdx.x] = s;
}

// ---------------- kernel 3: DFL partial sums (4 anchors / thread) ----------------
__global__ void __launch_bounds__(256)
dfl_partial_kernel(const float* __restrict__ pred_reg,     // [B,4,BINS,N]
                   const float* __restrict__ pseudo_dfl,   // [B,4,N]
                   const unsigned char* __restrict__ pos_mask,
                   float* __restrict__ ws_dfl) {
  __shared__ float sred[8];
  const int b = blockIdx.y;
  const int SN4 = SIDES_ * N4_;  // 8400 vec4 anchors per batch
  float acc = 0.0f;
  for (int v = blockIdx.x * 256 + threadIdx.x; v < SN4; v += DCH_ * 256) {
    const int s  = v / N4_;
    const int n4 = v - s * N4_;
    const int n  = n4 * 4;

    const float* __restrict__ rp =
        pred_reg + ((size_t)(b * SIDES_ + s) * BINS_) * N_ + n;

    v4f r[BINS_];
#pragma unroll
    for (int k = 0; k < BINS_; ++k)
      r[k] = __builtin_nontemporal_load(
                 reinterpret_cast<const v4f*>(rp + (size_t)k * N_));  // b128, coalesced

    v4f td4 = __builtin_nontemporal_load(
        reinterpret_cast<const v4f*>(pseudo_dfl + (size_t)(b * SIDES_ + s) * N_ + n));

    const uchar4 pm4 = *reinterpret_cast<const uchar4*>(pos_mask + (size_t)b * N_ + n);
    const float pmf[4] = {(float)pm4.x, (float)pm4.y, (float)pm4.z, (float)pm4.w};

#pragma unroll
    for (int j = 0; j < 4; ++j) {   // 4 independent log-softmax pipelines
      float mx = r[0][j];
#pragma unroll
      for (int k = 1; k < BINS_; ++k) mx = fmaxf(mx, r[k][j]);

      float td  = td4[j];
      float tlf = floorf(td);
      int   tl  = (int)tlf;
      int   tr  = tl + 1;
      float wr  = td - tlf;
      float wl  = 1.0f - wr;

      float se = 0.0f, rl = 0.0f, rr = 0.0f;
#pragma unroll
      for (int k = 0; k < BINS_; ++k) {
        se += __expf(r[k][j] - mx);
        rl = (k == tl) ? r[k][j] : rl;
        rr = (k == tr) ? r[k][j] : rr;
      }
      float lse = __logf(se);
      // ce = (wl+wr)*(mx+lse) - wl*rl - wr*rr, with wl+wr == 1
      acc += pmf[j] * (mx + lse - (wl * rl + wr * rr));
    }
  }
  float s = block_reduce_256(acc, sred);
  if (threadIdx.x == 0) ws_dfl[b * DCH_ + blockIdx.x] = s;
}

// ---------------- kernel 4: combine (one wave32 block) ----------------
__global__ void __launch_bounds__(32)
combine_kernel(const float* __restrict__ ws_pos,
               const float* __restrict__ ws_qfl,
               const float* __restrict__ ws_dfl,
               float* __restrict__ out) {
  const int b = threadIdx.x;   // 32 lanes = 32 batches, EXEC all ones
  float pos = ws_pos[b];
  float q = 0.0f;
  for (int i = 0; i < QCH_; ++i) q += ws_qfl[b * QCH_ + i];
  float d = 0.0f;
  for (int i = 0; i < DCH_; ++i) d += ws_dfl[b * DCH_ + i];
  float qv = q / (pos + EPS_);
  float dv = d / (4.0f * pos + EPS_);
  float qs = wave_reduce_wmma(qv, 0.0f);
  float ds = wave_reduce_wmma(dv, 0.0f);
  if (b == 0) {
    out[0] = qs * (1.0f / (float)B_);
    out[1] = ds * (1.0f / (float)B_);
  }
}

// ---------------- host launcher ----------------
extern "C" void kernel_launch(void* const* d_in, const int* in_sizes, int n_in,
                              void* d_out, int out_size, void* d_ws, size_t ws_size,
                              hipStream_t stream) {
  (void)in_sizes; (void)n_in; (void)out_size; (void)ws_size;
  const float*         pseudo_qfl = (const float*)d_in[0];          // [B,N,C]
  const float*         pseudo_dfl = (const float*)d_in[1];          // [B,4,N]
  const float*         pred_cls   = (const float*)d_in[2];          // [B,N,C]
  const float*         pred_reg   = (const float*)d_in[3];          // [B,4,BINS,N]
  const unsigned char* pos_mask   = (const unsigned char*)d_in[4];  // [B,N] bool
  const float*         total_mask = (const float*)d_in[5];          // [B,N,C]
  float* out = (float*)d_out;                                       // (qfl, dfl)

  float* ws_pos = (float*)d_ws;                 // 32
  float* ws_qfl = ws_pos + B_;                  // 32*96
  float* ws_dfl = ws_qfl + B_ * QCH_;           // 32*33

  pos_sum_kernel<<<B_, 256, 0, stream>>>(pos_mask, ws_pos);
  qfl_partial_kernel<<<dim3(QCH_, B_), 256, 0, stream>>>(pred_cls, pseudo_qfl,
                                                         total_mask, ws_qfl);
  dfl_partial_kernel<<<dim3(DCH_, B_), 256, 0, stream>>>(pred_reg, pseudo_dfl,
                                                         pos_mask, ws_dfl);
  combine_kernel<<<1, 32, 0, stream>>>(ws_pos, ws_qfl, ws_dfl, out);
}